// StructureProbe_43379169690301
// MI455X (gfx1250) — compile-verified
//
#include <hip/hip_runtime.h>

typedef __attribute__((ext_vector_type(2))) float v2f;
typedef __attribute__((ext_vector_type(8))) float v8f;

#define BATCH 4
#define SEQ   1024
#define NIN   1024
#define RANK  64

// ---------------------------------------------------------------------------
// Kernel 1: T[4096,64] = x[4096,1024] @ proj[1024,64], fp32 WMMA 16x16x4.
// One wave per 16x16 tile of T. 256 M-tiles x 4 N-tiles = 1024 waves.
// ---------------------------------------------------------------------------
__global__ __launch_bounds__(256) void proj_kernel(const float* __restrict__ x,
                                                   const float* __restrict__ p,
                                                   float* __restrict__ t) {
    const int wave = (blockIdx.x * blockDim.x + threadIdx.x) >> 5;
    const int lane = threadIdx.x & 31;
    const int mt   = wave >> 2;                 // 0..255
    const int nt   = wave & 3;                  // 0..3
    const int l15  = lane & 15;
    const int m    = (mt << 4) + l15;           // A row for this lane
    const int n    = (nt << 4) + l15;           // B col for this lane
    const int kk   = (lane >> 4) << 1;          // 0 or 2 (K sub-pair)

    const float* __restrict__ xrow = x + (size_t)m * NIN;

    v8f acc = {};
#pragma unroll 8
    for (int k0 = 0; k0 < NIN; k0 += 4) {
        // A: 16x4 fp32 tile of x (consecutive pair -> b64 load)
        v2f a = *(const v2f*)(xrow + k0 + kk);
        // B: 4x16 fp32 tile of proj (row-major [NIN,RANK], stride RANK)
        v2f b;
        b.x = p[(size_t)(k0 + kk) * RANK + n];
        b.y = p[(size_t)(k0 + kk + 1) * RANK + n];
        acc = __builtin_amdgcn_wmma_f32_16x16x4_f32(false, a, false, b,
                                                    (short)0, acc, false, false);
    }

    // C/D layout: VGPR v -> row (v + (lane<16?0:8)), col = l15
    const int mb   = mt << 4;
    const int rofs = (lane >> 4) << 3;          // 0 or 8
#pragma unroll
    for (int v = 0; v < 8; ++v) {
        t[(size_t)(mb + rofs + v) * RANK + n] = acc[v];
    }
}

// ---------------------------------------------------------------------------
// Kernel 2: nrm[row] = sum_r T[row][r]^2, one thread per row (4096 rows).
// ---------------------------------------------------------------------------
__global__ __launch_bounds__(256) void norm_kernel(const float* __restrict__ t,
                                                   float* __restrict__ nrm) {
    const int row = blockIdx.x * blockDim.x + threadIdx.x;  // 0..4095
    const float* __restrict__ tr = t + (size_t)row * RANK;
    float s = 0.0f;
#pragma unroll
    for (int r = 0; r < RANK; r += 2) {
        v2f v = *(const v2f*)(tr + r);
        s += v.x * v.x + v.y * v.y;
    }
    nrm[row] = s;
}

// ---------------------------------------------------------------------------
// Kernel 3: D[b,i,j] = nrm_i + nrm_j - 2 * (T_i . T_j), Gram via WMMA.
// One wave per 16x16 output tile; 4 batches * 64 * 64 tiles = 16384 waves.
// ---------------------------------------------------------------------------
__global__ __launch_bounds__(256) void dist_kernel(const float* __restrict__ t,
                                                   const float* __restrict__ nrm,
                                                   float* __restrict__ out) {
    const int wave = (blockIdx.x * blockDim.x + threadIdx.x) >> 5;
    const int lane = threadIdx.x & 31;
    const int b    = wave >> 12;                // 4096 tiles per batch
    const int tile = wave & 4095;
    const int it   = tile >> 6;                 // 0..63 (i tile)
    const int jt   = tile & 63;                 // 0..63 (j tile)
    const int l15  = lane & 15;
    const int kk   = (lane >> 4) << 1;          // 0 or 2

    const float* __restrict__ tb = t + (size_t)b * SEQ * RANK;
    const float* __restrict__ ti = tb + ((size_t)((it << 4) + l15)) * RANK;
    const float* __restrict__ tj = tb + ((size_t)((jt << 4) + l15)) * RANK;

    v8f acc = {};
#pragma unroll
    for (int k0 = 0; k0 < RANK; k0 += 4) {
        v2f a = *(const v2f*)(ti + k0 + kk);    // A = T rows (i side)
        v2f bb = *(const v2f*)(tj + k0 + kk);   // B = T^T cols (j side)
        acc = __builtin_amdgcn_wmma_f32_16x16x4_f32(false, a, false, bb,
                                                    (short)0, acc, false, false);
    }

    const float* __restrict__ nb = nrm + b * SEQ;
    const float nj = nb[(jt << 4) + l15];       // col norm for this lane
    const int   rofs = (lane >> 4) << 3;        // 0 or 8
    float* __restrict__ ob = out + (size_t)b * SEQ * SEQ;

#pragma unroll
    for (int v = 0; v < 8; ++v) {
        const int ig = (it << 4) + rofs + v;    // global row
        const float d = nb[ig] + nj - 2.0f * acc[v];
        ob[(size_t)ig * SEQ + (jt << 4) + l15] = d;
    }
}

// ---------------------------------------------------------------------------
extern "C" void kernel_launch(void* const* d_in, const int* in_sizes, int n_in,
                              void* d_out, int out_size, void* d_ws, size_t ws_size,
                              hipStream_t stream) {
    const float* x    = (const float*)d_in[0];   // [4,1024,1024]
    const float* proj = (const float*)d_in[1];   // [1024,64]
    float*       out  = (float*)d_out;           // [4,1024,1024]

    float* T   = (float*)d_ws;                   // [4096,64]  (1 MB)
    float* nrm = T + (size_t)BATCH * SEQ * RANK; // [4096]     (16 KB)

    // 1024 waves, 8 waves/block -> 128 blocks
    proj_kernel<<<128, 256, 0, stream>>>(x, proj, T);
    // 4096 threads -> 16 blocks
    norm_kernel<<<16, 256, 0, stream>>>(T, nrm);
    // 16384 waves, 8 waves/block -> 2048 blocks
    dist_kernel<<<2048, 256, 0, stream>>>(T, nrm, out);
}